// ACTR_80023830659519
// MI455X (gfx1250) — compile-verified
//
#include <hip/hip_runtime.h>
#include <hip/hip_bf16.h>

typedef float v2f __attribute__((ext_vector_type(2)));
typedef float v8f __attribute__((ext_vector_type(8)));

constexpr int   Bc    = 16384;   // batch
constexpr int   Dc    = 64;      // embedding dim
constexpr float GAMMA = 0.5f;

__device__ __forceinline__ float wsum32(float x) {
#pragma unroll
  for (int off = 16; off > 0; off >>= 1) x += __shfl_xor(x, off, 32);
  return x;
}

__device__ __forceinline__ float neg_logsig(float x) {
  // -log(sigmoid(x)) = softplus(-x), numerically stable
  return fmaxf(-x, 0.0f) + log1pf(expf(-fabsf(x)));
}

// ---------------------------------------------------------------------------
// Kernel 1: attention GEMM via fp32 WMMA (16x16x4), gather fused into A-loads.
// Row layout: r = t*B + b, t=0:user, t=1:anchor item, t=2..5:anchor meta g.
// Output: dots[r] = (X_row @ att_w_W + att_w_b) . att_v_W[sel_half]
// ---------------------------------------------------------------------------
__global__ __launch_bounds__(256) void actr_att_wmma(
    const int* __restrict__ u_id, const int* __restrict__ anchor,
    const int* __restrict__ item_meta,
    const float* __restrict__ user_emb, const float* __restrict__ item_emb,
    const float* __restrict__ meta_emb,
    const float* __restrict__ attW, const float* __restrict__ attWb,
    const float* __restrict__ attV,
    float* __restrict__ dots) {
  __shared__ float Wlds[64 * 65];        // row-padded to 65 to avoid conflicts
  __shared__ float red[8][16][17];       // per-wave partial row-dot buffer

  const int tid = threadIdx.x;
  for (int i = tid; i < 64 * 64; i += 256)
    Wlds[(i >> 6) * 65 + (i & 63)] = attW[i];
  __syncthreads();

  const int wave = tid >> 5;
  const int l    = tid & 31;
  const int lo   = l & 15;
  const int hi   = l >> 4;
  const int koff = hi * 2;               // K offset within 4-wide K slice

  const int tile = blockIdx.x * 8 + wave;      // 6144 tiles total
  const int r0   = tile * 16;
  const int t    = r0 / Bc;                    // row type (wave-uniform)
  const int b    = (r0 % Bc) + lo;             // batch element for this row

  const float* src;
  if (t == 0)      src = user_emb + (size_t)u_id[b] * Dc;
  else if (t == 1) src = item_emb + (size_t)anchor[b] * Dc;
  else {
    const int mi = item_meta[(size_t)anchor[b] * 4 + (t - 2)];
    src = meta_emb + (size_t)mi * Dc;
  }

  v8f c[4];
#pragma unroll
  for (int nt = 0; nt < 4; ++nt) c[nt] = (v8f){0, 0, 0, 0, 0, 0, 0, 0};

#pragma unroll
  for (int k0 = 0; k0 < 64; k0 += 4) {
    // A fragment: 16x4 fp32, lanes 0-15 K={k0,k0+1}, lanes 16-31 K={k0+2,k0+3}
    v2f a = *(const v2f*)(src + k0 + koff);
#pragma unroll
    for (int nt = 0; nt < 4; ++nt) {
      const int N = nt * 16 + lo;
      v2f bb;
      bb[0] = Wlds[(k0 + koff) * 65 + N];
      bb[1] = Wlds[(k0 + koff + 1) * 65 + N];
      c[nt] = __builtin_amdgcn_wmma_f32_16x16x4_f32(
          false, a, false, bb, (short)0, c[nt], false, false);
    }
  }

  // select att_v_W half: t==0 -> wu half [0:64), else wic half [64:128)
  const int base = (t == 0) ? 0 : 64;
  float vs[4];
#pragma unroll
  for (int nt = 0; nt < 4; ++nt) vs[nt] = attV[base + nt * 16 + lo];

  // bias contribution: att_w_b . att_v_half (wave-uniform)
  float pw = attWb[l] * attV[base + l] + attWb[l + 32] * attV[base + l + 32];
  pw = wsum32(pw);

  // contract Y tile against vs: C/D layout M = j + 8*hi, N = nt*16 + lo
#pragma unroll
  for (int j = 0; j < 8; ++j) {
    float q = c[0][j] * vs[0] + c[1][j] * vs[1] + c[2][j] * vs[2] + c[3][j] * vs[3];
    red[wave][j + 8 * hi][lo] = q;
  }
  __syncthreads();
  if (l < 16) {
    float s = 0.f;
#pragma unroll
    for (int cI = 0; cI < 16; ++cI) s += red[wave][l][cI];
    dots[r0 + l] = s + pw;
  }
}

// ---------------------------------------------------------------------------
// Kernel 2: one wave32 per batch element; lane = float2 slice of D.
// ---------------------------------------------------------------------------
__global__ __launch_bounds__(256) void actr_main(
    const int* __restrict__ u_id, const int* __restrict__ anchor,
    const int* __restrict__ pos_r, const int* __restrict__ pos_i,
    const int* __restrict__ neg_r, const int* __restrict__ neg_i,
    const int* __restrict__ neg_ri, const int* __restrict__ item_meta,
    const float* __restrict__ user_emb, const float* __restrict__ rel_emb,
    const float* __restrict__ item_emb, const float* __restrict__ item_emb_r,
    const float* __restrict__ item_bias, const float* __restrict__ rel_bias,
    const float* __restrict__ meta_emb, const float* __restrict__ att_v_b,
    const float* __restrict__ dots, float* __restrict__ blockSums) {
  __shared__ float acc3[3][8];
  const int tid  = threadIdx.x;
  const int wave = tid >> 5;
  const int l    = tid & 31;
  const int b    = blockIdx.x * 8 + wave;

  const int uid  = u_id[b];
  const int ai   = anchor[b];
  const int pri  = pos_r[b];
  const int pii  = pos_i[b];
  const int nrr  = neg_r[b];
  const int nii  = neg_i[b];
  const int nrii = neg_ri[b];

  auto row2 = [&](const float* basep, int row) -> float2 {
    return ((const float2*)(basep + (size_t)row * Dc))[l];
  };

  const float2 u2  = row2(user_emb, uid);
  const float2 a2  = row2(item_emb, ai);
  const float2 ar2 = row2(item_emb_r, ai);
  const float2 p2  = row2(item_emb, pii);
  const float2 n2  = row2(item_emb, nii);
  const float2 x2  = row2(item_emb, nrii);
  float2 rl[3];
#pragma unroll
  for (int r = 0; r < 3; ++r) rl[r] = row2(rel_emb, r);

  float2 am[4], pm[4], nm[4];
#pragma unroll
  for (int g = 0; g < 4; ++g) {
    am[g] = row2(meta_emb, item_meta[(size_t)ai * 4 + g]);
    pm[g] = row2(meta_emb, item_meta[(size_t)pii * 4 + g]);
    nm[g] = row2(meta_emb, item_meta[(size_t)nii * 4 + g]);
  }

  // predict_relation: pred[r] = rel_bias[r] - ||u + a_r - rel[r]||^2
  float pred[3];
#pragma unroll
  for (int r = 0; r < 3; ++r) {
    const float dx = u2.x + ar2.x - rl[r].x;
    const float dy = u2.y + ar2.y - rl[r].y;
    pred[r] = wsum32(dx * dx + dy * dy);
  }
#pragma unroll
  for (int r = 0; r < 3; ++r) pred[r] = rel_bias[r] - pred[r];
  const float m3 = fmaxf(pred[0], fmaxf(pred[1], pred[2]));
  const float e0 = expf(pred[0] - m3), e1 = expf(pred[1] - m3), e2 = expf(pred[2] - m3);
  const float inv3 = 1.0f / (e0 + e1 + e2);
  const float rw[3] = {e0 * inv3, e1 * inv3, e2 * inv3};
  const float relw[4] = {rw[0] * GAMMA, rw[1] * GAMMA, rw[2] * GAMMA, 1.0f - GAMMA};

  // attention coefficients from precomputed dots
  const float vb  = att_v_b[0];
  const float dv0 = dots[b];
  float vg[5];
#pragma unroll
  for (int g = 0; g < 5; ++g) vg[g] = dv0 + dots[(g + 1) * Bc + b] + vb;
  float m5 = vg[0];
#pragma unroll
  for (int g = 1; g < 5; ++g) m5 = fmaxf(m5, vg[g]);
  float es = 0.f, ce[5];
#pragma unroll
  for (int g = 0; g < 5; ++g) { ce[g] = expf(vg[g] - m5); es += ce[g]; }
  const float einv = 1.0f / es;
  float coef[5];
#pragma unroll
  for (int g = 0; g < 5; ++g) coef[g] = ce[g] * einv;

  // predict_score(pos) and predict_score(neg)
  const float2 ip[5] = {a2, am[0], am[1], am[2], am[3]};
  const float2 jp[5] = {p2, pm[0], pm[1], pm[2], pm[3]};
  const float2 jn[5] = {n2, nm[0], nm[1], nm[2], nm[3]};
  float sp = 0.f, sn = 0.f;
#pragma unroll
  for (int r = 0; r < 4; ++r) {
    const float2 ru = (r < 3) ? rl[r] : u2;  // rue = [rel_emb rows, u]
    float trp = 0.f, trn = 0.f;
#pragma unroll
    for (int g = 0; g < 5; ++g) {
      const float px = ip[g].x + ru.x - jp[g].x;
      const float py = ip[g].y + ru.y - jp[g].y;
      trp += coef[g] * (px * px + py * py);
      const float qx = ip[g].x + ru.x - jn[g].x;
      const float qy = ip[g].y + ru.y - jn[g].y;
      trn += coef[g] * (qx * qx + qy * qy);
    }
    sp += relw[r] * trp;
    sn += relw[r] * trn;
  }

  // item loss distances
  const float2 iir = (pri == 0) ? rl[0] : ((pri == 1) ? rl[1] : rl[2]);
  const float tpx = a2.x + iir.x - p2.x, tpy = a2.y + iir.y - p2.y;
  const float tnx = a2.x + iir.x - x2.x, tny = a2.y + iir.y - x2.y;
  float tp = tpx * tpx + tpy * tpy;
  float tn = tnx * tnx + tny * tny;

  sp = wsum32(sp); sn = wsum32(sn); tp = wsum32(tp); tn = wsum32(tn);

  const float bpos = item_bias[pii];
  const float posR = bpos - sp;
  const float negR = item_bias[nii] - sn;
  const float seqterm = neg_logsig(posR - negR);

  const float rpos = (pri == 0) ? rw[0] : ((pri == 1) ? rw[1] : rw[2]);
  const float rneg = (nrr == 0) ? rw[0] : ((nrr == 1) ? rw[1] : rw[2]);
  const float relterm = neg_logsig(rpos - rneg);

  const float pis = bpos - tp;
  const float nis = item_bias[nrii] - tn;
  const float itemterm = neg_logsig(pis - nis);

  if (l == 0) {
    acc3[0][wave] = seqterm;
    acc3[1][wave] = relterm;
    acc3[2][wave] = itemterm;
  }
  __syncthreads();
  if (tid < 3) {
    float s = 0.f;
#pragma unroll
    for (int w = 0; w < 8; ++w) s += acc3[tid][w];
    blockSums[blockIdx.x * 3 + tid] = s;
  }
}

// ---------------------------------------------------------------------------
// Kernel 3: deterministic final reduction -> 4 scalars
// ---------------------------------------------------------------------------
__global__ __launch_bounds__(256) void actr_finalize(
    const float* __restrict__ blockSums, int nblocks, float* __restrict__ out) {
  __shared__ float red[3][256];
  const int t = threadIdx.x;
  float s0 = 0.f, s1 = 0.f, s2 = 0.f;
  for (int j = t; j < nblocks; j += 256) {
    s0 += blockSums[j * 3 + 0];
    s1 += blockSums[j * 3 + 1];
    s2 += blockSums[j * 3 + 2];
  }
  red[0][t] = s0; red[1][t] = s1; red[2][t] = s2;
  __syncthreads();
  if (t == 0) {
    float a0 = 0.f, a1 = 0.f, a2 = 0.f;
    for (int i = 0; i < 256; ++i) { a0 += red[0][i]; a1 += red[1][i]; a2 += red[2][i]; }
    const float invB = 1.0f / (float)Bc;
    const float seq = a0 * invB, rlo = a1 * invB, ilo = a2 * invB;
    out[0] = seq + rlo + ilo;  // loss (ALPHA=BETA=1)
    out[1] = rlo;              // relation_loss
    out[2] = seq;              // seq_loss
    out[3] = ilo;              // item_loss
  }
}

extern "C" void kernel_launch(void* const* d_in, const int* in_sizes, int n_in,
                              void* d_out, int out_size, void* d_ws, size_t ws_size,
                              hipStream_t stream) {
  const int*   u_id       = (const int*)d_in[0];
  const int*   anchor     = (const int*)d_in[1];
  const int*   pos_r      = (const int*)d_in[2];
  const int*   pos_i      = (const int*)d_in[3];
  const int*   neg_r      = (const int*)d_in[4];
  const int*   neg_i      = (const int*)d_in[5];
  const int*   neg_ri     = (const int*)d_in[6];
  const int*   item_meta  = (const int*)d_in[7];
  const float* user_emb   = (const float*)d_in[8];
  const float* rel_emb    = (const float*)d_in[9];
  const float* item_emb   = (const float*)d_in[10];
  const float* item_emb_r = (const float*)d_in[11];
  const float* item_bias  = (const float*)d_in[12];
  const float* rel_bias   = (const float*)d_in[13];
  const float* meta_emb   = (const float*)d_in[14];
  const float* att_w_W    = (const float*)d_in[15];
  const float* att_w_b    = (const float*)d_in[16];
  const float* att_v_W    = (const float*)d_in[17];
  const float* att_v_b    = (const float*)d_in[18];

  float* dots      = (float*)d_ws;           // 6*B floats
  float* blockSums = dots + 6 * Bc;          // 2048*3 floats

  // 6*B/16 = 6144 row tiles, 8 waves (tiles) per 256-thread block -> 768 blocks
  actr_att_wmma<<<768, 256, 0, stream>>>(u_id, anchor, item_meta,
                                         user_emb, item_emb, meta_emb,
                                         att_w_W, att_w_b, att_v_W, dots);

  // B/8 = 2048 blocks, one wave32 per batch element
  actr_main<<<2048, 256, 0, stream>>>(u_id, anchor, pos_r, pos_i, neg_r, neg_i,
                                      neg_ri, item_meta, user_emb, rel_emb,
                                      item_emb, item_emb_r, item_bias, rel_bias,
                                      meta_emb, att_v_b, dots, blockSums);

  actr_finalize<<<1, 256, 0, stream>>>(blockSums, 2048, (float*)d_out);
}